// SequenceBlock_5506148074014
// MI455X (gfx1250) — compile-verified
//
#include <hip/hip_runtime.h>
#include <hip/hip_bf16.h>

typedef float v2f __attribute__((ext_vector_type(2)));
typedef float v8f __attribute__((ext_vector_type(8)));

// V_WMMA_F32_16X16X4_F32: D = A(16x4) x B(4x16) + C(16x16), f32, wave32.
#define WMMA4(A_, B_, C_) \
  __builtin_amdgcn_wmma_f32_16x16x4_f32(false, (A_), false, (B_), (short)0, (C_), false, false)

__device__ __forceinline__ void cmul(float ar, float ai, float br, float bi,
                                     float& cr, float& ci) {
  cr = ar * br - ai * bi;
  ci = ar * bi + ai * br;
}
__device__ __forceinline__ void cinv(float ar, float ai, float& cr, float& ci) {
  float id = 1.f / (ar * ar + ai * ai);
  cr = ar * id;
  ci = -ai * id;
}

// ---- CDNA5 async global->LDS data movers (ASYNCcnt-tracked), inline asm ----
// Generic pointers into LDS carry the wave-relative LDS offset in the low 32
// bits, which is exactly what VDST of GLOBAL_LOAD_ASYNC_TO_LDS expects.
__device__ __forceinline__ void async_g2l_b128(void* lds, const void* gptr) {
  unsigned int l = (unsigned int)(unsigned long long)lds;
  unsigned long long g = (unsigned long long)gptr;
  asm volatile("global_load_async_to_lds_b128 %0, %1, off"
               :: "v"(l), "v"(g)
               : "memory");
}
__device__ __forceinline__ void async_g2l_b32(void* lds, const void* gptr) {
  unsigned int l = (unsigned int)(unsigned long long)lds;
  unsigned long long g = (unsigned long long)gptr;
  asm volatile("global_load_async_to_lds_b32 %0, %1, off"
               :: "v"(l), "v"(g)
               : "memory");
}
__device__ __forceinline__ void wait_async0() {
  asm volatile("s_wait_asynccnt 0" ::: "memory");
}

// ---------------------------------------------------------------- LayerNorm
__global__ __launch_bounds__(256) void k_ln(const float* __restrict__ x,
                                            const float* __restrict__ nw,
                                            const float* __restrict__ nb,
                                            float* __restrict__ u) {
  int t = blockIdx.x, tid = threadIdx.x;
  __shared__ float red[256];
  float v0 = x[t * 512 + tid];
  float v1 = x[t * 512 + 256 + tid];
  red[tid] = v0 + v1;
  __syncthreads();
  for (int s = 128; s > 0; s >>= 1) {
    if (tid < s) red[tid] += red[tid + s];
    __syncthreads();
  }
  float mu = red[0] * (1.f / 512.f);
  __syncthreads();
  float d0 = v0 - mu, d1 = v1 - mu;
  red[tid] = d0 * d0 + d1 * d1;
  __syncthreads();
  for (int s = 128; s > 0; s >>= 1) {
    if (tid < s) red[tid] += red[tid + s];
    __syncthreads();
  }
  float inv = rsqrtf(red[0] * (1.f / 512.f) + 1e-5f);
  u[t * 512 + tid] = d0 * inv * nw[tid] + nb[tid];
  u[t * 512 + 256 + tid] = d1 * inv * nw[256 + tid] + nb[256 + tid];
}

// --------------------------------------------- Discretize (closed form, rank-1)
// Ab[i][j] = d_i*(delta_ij*(2/dt+lam_i)) - d_i*p_i*(conj(p_j) + s*w_j)
// w_j = conj(p_j)*(d_j*(2/dt+lam_j) - sigma);  sigma = sum_k |p_k|^2 d_k
// Also emits S0 = [Bb, Ab Bb, ..., Ab^15 Bb]  (16 columns).
__global__ __launch_bounds__(64) void k_disc(
    const float* __restrict__ Lre, const float* __restrict__ Lim,
    const float* __restrict__ Pre, const float* __restrict__ Pim,
    const float* __restrict__ Bre, const float* __restrict__ Bim,
    const float* __restrict__ lstep, float* __restrict__ Ab,
    float* __restrict__ S0) {
  int h = blockIdx.x, i = threadIdx.x;
  __shared__ float pre_[64], pim_[64], wre[64], wim[64];
  __shared__ float tr_[64], ti_[64];
  __shared__ float AbRe[4096], AbIm[4096];
  __shared__ float Vr[64], Vi[64];
  __shared__ float sred[4];

  float step = __expf(lstep[h]);
  float two = 2.f / step;
  float lr = Lre[h * 64 + i], li = Lim[h * 64 + i];
  float pr = Pre[h * 64 + i], pi = Pim[h * 64 + i];
  pre_[i] = pr;
  pim_[i] = pi;
  float dr, di;
  cinv(two - lr, -li, dr, di);  // d = 1/(2/dt - lam)
  float p2 = pr * pr + pi * pi;
  tr_[i] = p2 * dr;
  ti_[i] = p2 * di;
  __syncthreads();
  if (i == 0) {
    float sr = 0, si = 0;
    for (int k = 0; k < 64; ++k) { sr += tr_[k]; si += ti_[k]; }
    sred[0] = sr; sred[1] = si;
    float ivr, ivi;
    cinv(1.f + sr, si, ivr, ivi);  // s = 1/(1+sigma)
    sred[2] = ivr; sred[3] = ivi;
  }
  __syncthreads();
  float sgr = sred[0], sgi = sred[1], ssr = sred[2], ssi = sred[3];
  // w_j
  float t1r, t1i;
  cmul(dr, di, two + lr, li, t1r, t1i);
  t1r -= sgr; t1i -= sgi;
  float wjr, wji;
  cmul(pr, -pi, t1r, t1i, wjr, wji);
  wre[i] = wjr; wim[i] = wji;
  // beta = sum conj(p)*d*B
  float br_ = Bre[h * 64 + i], bi_ = Bim[h * 64 + i];
  float cpdr, cpdi;
  cmul(pr, -pi, dr, di, cpdr, cpdi);
  float tbr, tbi;
  cmul(cpdr, cpdi, br_, bi_, tbr, tbi);
  tr_[i] = tbr; ti_[i] = tbi;
  __syncthreads();
  if (i == 0) {
    float xr = 0, xi = 0;
    for (int k = 0; k < 64; ++k) { xr += tr_[k]; xi += ti_[k]; }
    sred[0] = xr; sred[1] = xi;
  }
  __syncthreads();
  float betar = sred[0], betai = sred[1];
  // Bb = 2*d*(B - s*p*beta)
  float sbr, sbi;
  cmul(ssr, ssi, betar, betai, sbr, sbi);
  float psr, psi;
  cmul(pr, pi, sbr, sbi, psr, psi);
  float Bbr, Bbi;
  cmul(dr, di, br_ - psr, bi_ - psi, Bbr, Bbi);
  Bbr *= 2.f; Bbi *= 2.f;
  // Ab row i
  float dpr, dpi;
  cmul(dr, di, pr, pi, dpr, dpi);
  for (int j = 0; j < 64; ++j) {
    float swr, swi;
    cmul(ssr, ssi, wre[j], wim[j], swr, swi);
    float qr = pre_[j] + swr, qi = -pim_[j] + swi;
    float er, ei;
    cmul(dpr, dpi, qr, qi, er, ei);
    float abr = -er, abi = -ei;
    if (j == i) {
      float ar2, ai2;
      cmul(dr, di, two + lr, li, ar2, ai2);
      abr += ar2; abi += ai2;
    }
    AbRe[i * 64 + j] = abr;
    AbIm[i * 64 + j] = abi;
    Ab[h * 8192 + i * 64 + j] = abr;
    Ab[h * 8192 + 4096 + i * 64 + j] = abi;
  }
  // S0 columns: V_{j+1} = Ab V_j, V_0 = Bb
  Vr[i] = Bbr; Vi[i] = Bbi;
  S0[h * 2048 + i * 16 + 0] = Bbr;
  S0[h * 2048 + 1024 + i * 16 + 0] = Bbi;
  __syncthreads();
  for (int j = 1; j < 16; ++j) {
    float ar = 0, ai = 0;
    for (int k = 0; k < 64; ++k) {
      float mr = AbRe[i * 64 + k], mi = AbIm[i * 64 + k];
      ar += mr * Vr[k] - mi * Vi[k];
      ai += mr * Vi[k] + mi * Vr[k];
    }
    __syncthreads();
    Vr[i] = ar; Vi[i] = ai;
    S0[h * 2048 + i * 16 + j] = ar;
    S0[h * 2048 + 1024 + i * 16 + j] = ai;
    __syncthreads();
  }
}

// ---------------------------- 64x64 complex matrix squaring via WMMA f32 16x16x4
// LDS staged with GLOBAL_LOAD_ASYNC_TO_LDS_B128 (ASYNCcnt), overlapped issue.
__global__ __launch_bounds__(256) void k_sq(const float* __restrict__ src,
                                            float* __restrict__ dst) {
  int h = blockIdx.x, tid = threadIdx.x;
  int wave = tid >> 5, lane = tid & 31, nl = lane & 15, half = lane >> 4;
  __shared__ __align__(16) float Mre[4096], Mim[4096];
  const float* s = src + h * 8192;
  for (int base = tid * 4; base < 4096; base += 1024) {
    async_g2l_b128(&Mre[base], s + base);
    async_g2l_b128(&Mim[base], s + 4096 + base);
  }
  wait_async0();
  __syncthreads();
  float* d = dst + h * 8192;
  for (int tt = 0; tt < 2; ++tt) {
    int tile = wave * 2 + tt;        // 16 complex 16x16 output tiles
    int rm = tile >> 2, rn = tile & 3;
    v8f rr = {}, ii = {}, ri = {}, ir = {};
    for (int kk = 0; kk < 16; ++kk) {
      int am = rm * 16 + nl, ak = kk * 4 + half * 2;
      v2f ar, ai, br, bi;
      ar.x = Mre[am * 64 + ak];       ar.y = Mre[am * 64 + ak + 1];
      ai.x = Mim[am * 64 + ak];       ai.y = Mim[am * 64 + ak + 1];
      int bk = kk * 4 + half * 2, bn = rn * 16 + nl;
      br.x = Mre[bk * 64 + bn];       br.y = Mre[(bk + 1) * 64 + bn];
      bi.x = Mim[bk * 64 + bn];       bi.y = Mim[(bk + 1) * 64 + bn];
      rr = WMMA4(ar, br, rr);
      ii = WMMA4(ai, bi, ii);
      ri = WMMA4(ar, bi, ri);
      ir = WMMA4(ai, br, ir);
    }
    for (int r = 0; r < 8; ++r) {
      int m = rm * 16 + r + half * 8, c = rn * 16 + nl;
      d[m * 64 + c] = rr[r] - ii[r];
      d[4096 + m * 64 + c] = ri[r] + ir[r];
    }
  }
}

// ------------- Cb: solve conj(I-AbL)^T y = C^T (Gauss-Jordan), Cb = conj(y)
__global__ __launch_bounds__(64) void k_inv(const float* __restrict__ AbL,
                                            const float* __restrict__ Cre,
                                            const float* __restrict__ Cim,
                                            float* __restrict__ Cb) {
  int h = blockIdx.x, i = threadIdx.x;
  __shared__ float Gr[64][65], Gi[64][65];
  __shared__ float cr[64], ci[64];
  for (int j = 0; j < 64; ++j) {
    float mr = AbL[h * 8192 + j * 64 + i];
    float mi = AbL[h * 8192 + 4096 + j * 64 + i];
    Gr[i][j] = (i == j ? 1.f : 0.f) - mr;  // conj(I - AbL)^T
    Gi[i][j] = mi;
  }
  cr[i] = Cre[h * 64 + i];
  ci[i] = Cim[h * 64 + i];
  __syncthreads();
  for (int k = 0; k < 64; ++k) {
    if (i == k) {
      float pr, pi;
      cinv(Gr[k][k], Gi[k][k], pr, pi);
      for (int j = 0; j < 64; ++j) {
        float a = Gr[k][j], b = Gi[k][j];
        Gr[k][j] = a * pr - b * pi;
        Gi[k][j] = a * pi + b * pr;
      }
      float a = cr[k], b = ci[k];
      cr[k] = a * pr - b * pi;
      ci[k] = a * pi + b * pr;
    }
    __syncthreads();
    if (i != k) {
      float fr = Gr[i][k], fi = Gi[i][k];
      for (int j = k; j < 64; ++j) {
        float a = Gr[k][j], b = Gi[k][j];
        Gr[i][j] -= fr * a - fi * b;
        Gi[i][j] -= fr * b + fi * a;
      }
      float a = cr[k], b = ci[k];
      cr[i] -= fr * a - fi * b;
      ci[i] -= fr * b + fi * a;
    }
    __syncthreads();
  }
  Cb[h * 64 + i] = cr[i];
  Cb[32768 + h * 64 + i] = -ci[i];
}

// ----- Chunked scan: emit conv-kernel taps K, accumulate hidden, S <- Ab^16 S
__global__ __launch_bounds__(64) void k_scan(const float* __restrict__ M16,
                                             const float* __restrict__ S0,
                                             const float* __restrict__ Cbp,
                                             const float* __restrict__ u,
                                             float* __restrict__ Kk,
                                             float* __restrict__ outHidden) {
  int h = blockIdx.x, tid = threadIdx.x;
  int wave = tid >> 5, lane = tid & 31, nl = lane & 15, half = lane >> 4;
  __shared__ __align__(16) float Mre[4096], Mim[4096];
  __shared__ __align__(16) float Sre[1024], Sim[1024];
  __shared__ float Cbr[64], Cbi[64], hidr[64], hidi[64];
  __shared__ float uch[16];
  // Async-stage Ab^16 (both planes) and S0 via the CDNA5 data mover.
  for (int base = tid * 4; base < 4096; base += 256) {
    async_g2l_b128(&Mre[base], M16 + h * 8192 + base);
    async_g2l_b128(&Mim[base], M16 + h * 8192 + 4096 + base);
  }
  for (int base = tid * 4; base < 1024; base += 256) {
    async_g2l_b128(&Sre[base], S0 + h * 2048 + base);
    async_g2l_b128(&Sim[base], S0 + h * 2048 + 1024 + base);
  }
  Cbr[tid] = Cbp[h * 64 + tid];
  Cbi[tid] = Cbp[32768 + h * 64 + tid];
  hidr[tid] = 0.f;
  hidi[tid] = 0.f;
  wait_async0();
  __syncthreads();
  for (int c = 0; c < 64; ++c) {
    if (tid < 16) {
      uch[tid] = u[(1023 - (c * 16 + tid)) * 512 + h];
      float acc = 0.f;  // K[16c+j] = Re(Cb . S[:,j])
      for (int k = 0; k < 64; ++k)
        acc += Cbr[k] * Sre[k * 16 + tid] - Cbi[k] * Sim[k * 16 + tid];
      Kk[h * 1024 + c * 16 + tid] = acc;
    }
    __syncthreads();
    {
      float hr = 0.f, hi = 0.f;  // hidden += S[:,j] * u[L-1-(16c+j)]
      for (int j = 0; j < 16; ++j) {
        float uv = uch[j];
        hr += Sre[tid * 16 + j] * uv;
        hi += Sim[tid * 16 + j] * uv;
      }
      hidr[tid] += hr;
      hidi[tid] += hi;
    }
    // S = M16 @ S  (64x64 x 64x16 complex) via 4 real WMMA accumulators
    v8f outr[2], outi[2];
    for (int tt = 0; tt < 2; ++tt) {
      int rm = wave + tt * 2;
      v8f rr = {}, ii = {}, ri = {}, ir = {};
      for (int kk = 0; kk < 16; ++kk) {
        int am = rm * 16 + nl, ak = kk * 4 + half * 2;
        v2f ar, ai, br, bi;
        ar.x = Mre[am * 64 + ak];       ar.y = Mre[am * 64 + ak + 1];
        ai.x = Mim[am * 64 + ak];       ai.y = Mim[am * 64 + ak + 1];
        int bk = kk * 4 + half * 2;
        br.x = Sre[bk * 16 + nl];       br.y = Sre[(bk + 1) * 16 + nl];
        bi.x = Sim[bk * 16 + nl];       bi.y = Sim[(bk + 1) * 16 + nl];
        rr = WMMA4(ar, br, rr);
        ii = WMMA4(ai, bi, ii);
        ri = WMMA4(ar, bi, ri);
        ir = WMMA4(ai, br, ir);
      }
      for (int r = 0; r < 8; ++r) {
        outr[tt][r] = rr[r] - ii[r];
        outi[tt][r] = ri[r] + ir[r];
      }
    }
    __syncthreads();
    for (int tt = 0; tt < 2; ++tt) {
      int rm = wave + tt * 2;
      for (int r = 0; r < 8; ++r) {
        int m = rm * 16 + r + half * 8;
        Sre[m * 16 + nl] = outr[tt][r];
        Sim[m * 16 + nl] = outi[tt][r];
      }
    }
    __syncthreads();
  }
  outHidden[(h * 64 + tid) * 2] = hidr[tid];
  outHidden[(h * 64 + tid) * 2 + 1] = hidi[tid];
}

// ------------------------------- Causal convolution + tanh-GELU (per head)
__global__ __launch_bounds__(256) void k_conv(const float* __restrict__ u,
                                              const float* __restrict__ Kk,
                                              const float* __restrict__ Dp,
                                              float* __restrict__ g) {
  int h = blockIdx.x, tid = threadIdx.x;
  __shared__ __align__(16) float Kh[1024];
  __shared__ float uh[1024];
  for (int base = tid * 4; base < 1024; base += 1024)
    async_g2l_b128(&Kh[base], Kk + h * 1024 + base);
  for (int i = tid; i < 1024; i += 256) uh[i] = u[i * 512 + h];
  wait_async0();
  __syncthreads();
  float Dh = Dp[h];
  for (int t = tid; t < 1024; t += 256) {
    float acc = Dh * uh[t];
    for (int m = 0; m <= t; ++m) acc += Kh[m] * uh[t - m];
    float x3 = acc * acc * acc;
    g[t * 512 + h] =
        0.5f * acc * (1.f + tanhf(0.7978845608028654f * (acc + 0.044715f * x3)));
  }
}

// -------- MLP: out = skip + (g W1^T + b1) * sigmoid(g W2^T + b2), WMMA GEMMs
// A tile async-staged with B128; W tiles transposed in-flight with per-lane
// B32 async transfers (arbitrary LDS destination per lane).
__global__ __launch_bounds__(256) void k_mlp(
    const float* __restrict__ g, const float* __restrict__ W1,
    const float* __restrict__ b1, const float* __restrict__ W2,
    const float* __restrict__ b2, const float* __restrict__ x,
    float* __restrict__ out) {
  int tb = blockIdx.x * 64, ob = blockIdx.y * 64;
  int tid = threadIdx.x, wave = tid >> 5, lane = tid & 31;
  int nl = lane & 15, half = lane >> 4;
  __shared__ __align__(16) float As[4096], B1s[4096], B2s[4096];
  v8f acc1[2] = {}, acc2[2] = {};
  for (int kc = 0; kc < 8; ++kc) {
    for (int base = tid * 4; base < 4096; base += 1024) {
      int r = base >> 6, cc = base & 63;
      async_g2l_b128(&As[base], g + (tb + r) * 512 + kc * 64 + cc);
    }
    for (int idx = tid; idx < 4096; idx += 256) {
      int r = idx >> 6, cc = idx & 63;  // B[k=r][n=cc] = W[o=ob+cc][k]
      async_g2l_b32(&B1s[idx], W1 + (ob + cc) * 512 + kc * 64 + r);
      async_g2l_b32(&B2s[idx], W2 + (ob + cc) * 512 + kc * 64 + r);
    }
    wait_async0();
    __syncthreads();
    for (int tt = 0; tt < 2; ++tt) {
      int tile = wave * 2 + tt;
      int rm = tile >> 2, rn = tile & 3;
      for (int kk = 0; kk < 16; ++kk) {
        v2f av, b1v, b2v;
        int am = rm * 16 + nl, ak = kk * 4 + half * 2;
        av.x = As[am * 64 + ak];
        av.y = As[am * 64 + ak + 1];
        int bk = kk * 4 + half * 2, bn = rn * 16 + nl;
        b1v.x = B1s[bk * 64 + bn];
        b1v.y = B1s[(bk + 1) * 64 + bn];
        b2v.x = B2s[bk * 64 + bn];
        b2v.y = B2s[(bk + 1) * 64 + bn];
        acc1[tt] = WMMA4(av, b1v, acc1[tt]);
        acc2[tt] = WMMA4(av, b2v, acc2[tt]);
      }
    }
    __syncthreads();
  }
  for (int tt = 0; tt < 2; ++tt) {
    int tile = wave * 2 + tt;
    int rm = tile >> 2, rn = tile & 3;
    for (int r = 0; r < 8; ++r) {
      int m = rm * 16 + r + half * 8, c = rn * 16 + nl;
      int t = tb + m, o = ob + c;
      float y1 = acc1[tt][r] + b1[o];
      float y2 = acc2[tt][r] + b2[o];
      float sig = 1.f / (1.f + __expf(-y2));
      out[t * 512 + o] = x[t * 512 + o] + y1 * sig;
    }
  }
}

extern "C" void kernel_launch(void* const* d_in, const int* in_sizes, int n_in,
                              void* d_out, int out_size, void* d_ws,
                              size_t ws_size, hipStream_t stream) {
  const float* x    = (const float*)d_in[0];
  const float* Lre  = (const float*)d_in[1];
  const float* Lim  = (const float*)d_in[2];
  const float* Pre  = (const float*)d_in[3];
  const float* Pim  = (const float*)d_in[4];
  const float* Bre  = (const float*)d_in[5];
  const float* Bim  = (const float*)d_in[6];
  const float* Cre  = (const float*)d_in[7];
  const float* Cim  = (const float*)d_in[8];
  const float* Dp   = (const float*)d_in[9];
  const float* lst  = (const float*)d_in[10];
  const float* nw   = (const float*)d_in[11];
  const float* nb   = (const float*)d_in[12];
  const float* W1   = (const float*)d_in[13];
  const float* b1   = (const float*)d_in[14];
  const float* W2   = (const float*)d_in[15];
  const float* b2   = (const float*)d_in[16];

  float* ws = (float*)d_ws;
  float* u  = ws;                 // 524288
  float* B0 = u + 524288;         // 4194304 (H x 64x64 complex, re/im planes)
  float* B1m = B0 + 4194304;      // 4194304
  float* B2m = B1m + 4194304;     // 4194304
  float* S0 = B2m + 4194304;      // 1048576
  float* Cb = S0 + 1048576;       // 65536
  float* Kk = Cb + 65536;         // 524288
  float* g  = Kk + 524288;        // 524288   (total ~61 MB)

  k_ln<<<1024, 256, 0, stream>>>(x, nw, nb, u);
  k_disc<<<512, 64, 0, stream>>>(Lre, Lim, Pre, Pim, Bre, Bim, lst, B0, S0);
  // Repeated squaring: Ab^2..Ab^1024; keep Ab^16 in B2m, Ab^1024 ends in B0.
  k_sq<<<512, 256, 0, stream>>>(B0, B1m);   // ^2
  k_sq<<<512, 256, 0, stream>>>(B1m, B2m);  // ^4
  k_sq<<<512, 256, 0, stream>>>(B2m, B1m);  // ^8
  k_sq<<<512, 256, 0, stream>>>(B1m, B2m);  // ^16  (kept)
  k_sq<<<512, 256, 0, stream>>>(B2m, B1m);  // ^32
  k_sq<<<512, 256, 0, stream>>>(B1m, B0);   // ^64
  k_sq<<<512, 256, 0, stream>>>(B0, B1m);   // ^128
  k_sq<<<512, 256, 0, stream>>>(B1m, B0);   // ^256
  k_sq<<<512, 256, 0, stream>>>(B0, B1m);   // ^512
  k_sq<<<512, 256, 0, stream>>>(B1m, B0);   // ^1024
  k_inv<<<512, 64, 0, stream>>>(B0, Cre, Cim, Cb);
  k_scan<<<512, 64, 0, stream>>>(B2m, S0, Cb, u, Kk, (float*)d_out);
  k_conv<<<512, 256, 0, stream>>>(u, Kk, Dp, g);
  dim3 mg(16, 8);
  k_mlp<<<mg, 256, 0, stream>>>(g, W1, b1, W2, b2, x, (float*)d_out + 65536);
}